// qgnn_59751585022674
// MI455X (gfx1250) — compile-verified
//
#include <hip/hip_runtime.h>

#define NN 100000
#define NE 3200000
#define D  64

typedef float v2f __attribute__((ext_vector_type(2)));
typedef float v8f __attribute__((ext_vector_type(8)));

// ---------------- Kernel 1: zero out accumulator + degree ----------------
__global__ void k_zero(float* __restrict__ out, float* __restrict__ deg) {
    int i = blockIdx.x * blockDim.x + threadIdx.x;
    if (i < NN * D) out[i] = 0.0f;
    if (i < NN)     deg[i] = 0.0f;
}

// ---------------- Kernel 2: degree histogram over col --------------------
__global__ void k_degree(const int* __restrict__ col, float* __restrict__ deg) {
    int e = blockIdx.x * blockDim.x + threadIdx.x;
    if (e < NE) atomicAdd(&deg[col[e]], 1.0f);
}

// ---------------- Kernel 3: dinv = deg>0 ? rsqrt(deg) : 0 ----------------
__global__ void k_dinv(const float* __restrict__ deg, float* __restrict__ dinv) {
    int i = blockIdx.x * blockDim.x + threadIdx.x;
    if (i < NN) {
        float d = deg[i];
        dinv[i] = (d > 0.0f) ? rsqrtf(d) : 0.0f;
    }
}

// ---------------- Kernel 4: h' = (x @ W) * dinv[row], fp32 WMMA ----------
// One block = 128 threads = 4 waves; block bi handles M-tile bi, wave w handles
// N-tile w (D=64 -> 4 N-tiles of 16). K=64 consumed in steps of 4 with
// V_WMMA_F32_16X16X4_F32. W (16 KB) is staged in LDS once per block.
__global__ __launch_bounds__(128) void k_gemm(const float* __restrict__ x,
                                              const float* __restrict__ W,
                                              const float* __restrict__ dinv,
                                              float* __restrict__ hp) {
    __shared__ float Wsh[D * D];
    const int tid = threadIdx.x;

    // cooperative load of W into LDS (1024 float4 / 128 threads = 8 each)
    {
        const float4* Wv   = (const float4*)W;
        float4*       Wshv = (float4*)Wsh;
        #pragma unroll
        for (int i = tid; i < (D * D) / 4; i += 128) Wshv[i] = Wv[i];
    }
    __syncthreads();

    const int wave = tid >> 5;        // N-tile index 0..3
    const int lane = tid & 31;
    const int half = lane >> 4;       // 0: K pair {0,1}; 1: K pair {2,3}
    const int l16  = lane & 15;
    const int mt   = blockIdx.x;      // M-tile (NN/16 = 6250 exact)
    const int mrow = mt * 16 + l16;   // A-matrix row for this lane
    const int colg = wave * 16 + l16; // global output column

    const float* xrow = x + (size_t)mrow * D;

    v8f c = {};
    #pragma unroll
    for (int k0 = 0; k0 < D; k0 += 4) {
        const int kk = k0 + half * 2;
        // A 16x4 fp32: lane<16 holds K={k0,k0+1}, lane>=16 holds K={k0+2,k0+3}
        v2f a = *(const v2f*)(xrow + kk);
        // B 4x16 fp32: row K striped across lanes within a VGPR
        v2f b;
        b.x = Wsh[kk * D + colg];
        b.y = Wsh[(kk + 1) * D + colg];
        // 8 args: (neg_a, A, neg_b, B, c_mod, C, reuse_a, reuse_b)
        c = __builtin_amdgcn_wmma_f32_16x16x4_f32(
                false, a, false, b, (short)0, c, false, false);
    }

    // C/D layout: VGPR r -> M = r (lanes 0-15) / r+8 (lanes 16-31), N = l16
    #pragma unroll
    for (int r = 0; r < 8; ++r) {
        const int m = mt * 16 + r + half * 8;
        hp[(size_t)m * D + colg] = c[r] * dinv[m];
    }
}

// ---------------- Kernel 5: scatter-add h'[col] into out[row] ------------
// 16 threads per edge, each moves one float4 (4 f32 atomics). Working set
// (hp + out + indices ~ 77 MB) fits in the 192 MB L2, so gathers/atomics
// resolve at L2 rather than HBM.
__global__ void k_scatter(const int* __restrict__ row,
                          const int* __restrict__ col,
                          const float* __restrict__ hp,
                          float* __restrict__ out) {
    long long gid = (long long)blockIdx.x * blockDim.x + threadIdx.x;
    if (gid >= (long long)NE * 16) return;
    const int e  = (int)(gid >> 4);
    const int d4 = (int)(gid & 15) << 2;
    const int r  = row[e];
    const int cc = col[e];
    const float4 v = *(const float4*)(hp + (size_t)cc * D + d4);
    float* o = out + (size_t)r * D + d4;
    atomicAdd(o + 0, v.x);
    atomicAdd(o + 1, v.y);
    atomicAdd(o + 2, v.z);
    atomicAdd(o + 3, v.w);
}

// ---------------- Kernel 6: out = out * dinv[node] + bias ----------------
__global__ void k_finalize(float* __restrict__ out,
                           const float* __restrict__ dinv,
                           const float* __restrict__ bias) {
    int i = blockIdx.x * blockDim.x + threadIdx.x;
    if (i < NN * D) {
        const int n = i >> 6;
        const int d = i & (D - 1);
        out[i] = out[i] * dinv[n] + bias[d];
    }
}

extern "C" void kernel_launch(void* const* d_in, const int* in_sizes, int n_in,
                              void* d_out, int out_size, void* d_ws, size_t ws_size,
                              hipStream_t stream) {
    const float* x          = (const float*)d_in[0];
    const int*   edge_index = (const int*)d_in[1];   // [2, NE] row-major
    const float* W          = (const float*)d_in[2];
    const float* bias       = (const float*)d_in[3];
    float*       out        = (float*)d_out;

    const int* row = edge_index;        // edge_index[0]
    const int* col = edge_index + NE;   // edge_index[1]

    // workspace layout
    char* ws   = (char*)d_ws;
    float* deg  = (float*)(ws);                       // NN floats (400 KB)
    float* dinv = (float*)(ws + (size_t)NN * 4);      // NN floats
    float* hp   = (float*)(ws + (size_t)2 * NN * 4);  // NN*D floats (25.6 MB)

    // 1) zero out + deg
    {
        int tot = NN * D;
        k_zero<<<(tot + 255) / 256, 256, 0, stream>>>(out, deg);
    }
    // 2) degree
    k_degree<<<(NE + 255) / 256, 256, 0, stream>>>(col, deg);
    // 3) dinv
    k_dinv<<<(NN + 255) / 256, 256, 0, stream>>>(deg, dinv);
    // 4) WMMA GEMM + col-side scaling
    k_gemm<<<NN / 16, 128, 0, stream>>>(x, W, dinv, hp);
    // 5) edge scatter-add
    {
        long long threads = (long long)NE * 16;
        int blocks = (int)((threads + 255) / 256);
        k_scatter<<<blocks, 256, 0, stream>>>(row, col, hp, out);
    }
    // 6) row-side scaling + bias
    k_finalize<<<(NN * D + 255) / 256, 256, 0, stream>>>(out, dinv, bias);
}